// VQBlock_10024453669118
// MI455X (gfx1250) — compile-verified
//
#include <hip/hip_runtime.h>
#include <hip/hip_bf16.h>

typedef __attribute__((ext_vector_type(2))) float v2f;
typedef __attribute__((ext_vector_type(8))) float v8f;

#define DIM   256     // embedding dim (reduction K of the GEMM)
#define KDIC  1024    // number of codes (N of the GEMM)
#define MT    32      // rows per workgroup
#define NT    64      // dictionary columns per LDS chunk
#define NCHUNK (KDIC / NT)   // 16
#define APAD  260     // LDS row stride (floats): 260%64=4 -> conflict-free frag reads, 16B-aligned rows
#define BPAD  260     // same for the dict tile (enables aligned b128 LDS stores)

// ---------------------------------------------------------------------------
// Kernel 0: transpose dictionary [256][1024] -> dictT [1024][256]
//           (makes the code-column gather and B staging fully coalesced)
// ---------------------------------------------------------------------------
__global__ __launch_bounds__(256) void vq_transpose(const float* __restrict__ dict,
                                                    float* __restrict__ dictT) {
    __shared__ float tile[32][33];
    const int k0 = blockIdx.x * 32;        // column block in dict
    const int d0 = blockIdx.y * 32;        // row block in dict
    const int tx = threadIdx.x & 31;
    const int ty = threadIdx.x >> 5;       // 0..7
    #pragma unroll
    for (int i = 0; i < 4; ++i) {
        int d = ty + i * 8;
        tile[d][tx] = dict[(long)(d0 + d) * KDIC + (k0 + tx)];   // coalesced read
    }
    __syncthreads();
    #pragma unroll
    for (int i = 0; i < 4; ++i) {
        int k = ty + i * 8;
        dictT[(long)(k0 + k) * DIM + (d0 + tx)] = tile[tx][k];   // coalesced write
    }
}

// ---------------------------------------------------------------------------
// Kernel 1: dictionary column squared norms  dnorm[k] = sum_d dict[d][k]^2
// ---------------------------------------------------------------------------
__global__ __launch_bounds__(256) void vq_norms(const float* __restrict__ dict,
                                                float* __restrict__ dnorm) {
    int k = blockIdx.x * 256 + threadIdx.x;   // 4 blocks * 256 = 1024
    float s = 0.f;
    #pragma unroll 4
    for (int d = 0; d < DIM; ++d) {
        float v = dict[(long)d * KDIC + k];   // coalesced across lanes
        s += v * v;
    }
    dnorm[k] = s;
}

// ---------------------------------------------------------------------------
// Kernel 2: main fused kernel.
//   Per workgroup: 32 rows of x. 8 waves arranged 2 (row tiles) x 4 (col tiles).
//   GEMM via v_wmma_f32_16x16x4_f32, argmin over 1024 codes, gather + loss.
//   dictT == nullptr selects the (slower) strided fallback paths.
// ---------------------------------------------------------------------------
__global__ __launch_bounds__(256) void vq_main(const float* __restrict__ x,
                                               const float* __restrict__ dict,
                                               const float* __restrict__ dictT,
                                               const float* __restrict__ dnorm,
                                               float* __restrict__ out,
                                               float* __restrict__ lossPartial) {
    __shared__ float As[MT][APAD];     // x rows   [m][k]
    __shared__ float Bs[NT][BPAD];     // dict cols[c][k] (column-major chunk)
    __shared__ float candS[MT][4];
    __shared__ int   candI[MT][4];
    __shared__ int   selIdx[MT];
    __shared__ float red[256];

    const int tid  = threadIdx.x;
    const int wave = tid >> 5;
    const int lane = tid & 31;
    const int half = lane >> 4;        // which 16-lane half of the wave
    const int l15  = lane & 15;
    const int wrow = (wave >> 2) << 4; // 0 or 16      : M offset of this wave's tile
    const int wcol = (wave & 3) << 4;  // 0,16,32,48   : N offset within the NT chunk
    const long m0  = (long)blockIdx.x * MT;
    const bool useT = (dictT != nullptr);

    // ---- stage 32 rows of x into LDS (float4, coalesced) -------------------
    {
        const float4* xv = (const float4*)(x + m0 * DIM);
        #pragma unroll
        for (int j = 0; j < 8; ++j) {
            int idx = tid + 256 * j;          // float4 index, 2048 total
            int row = idx >> 6;               // (idx*4)/256
            int d   = (idx & 63) << 2;
            float4 v = xv[idx];
            *(float4*)&As[row][d] = v;
        }
    }
    __syncthreads();

    // running per-lane argmin state: acc VGPR r <-> row (wrow + r + 8*half)
    float bestS[8];
    int   bestI[8];
    #pragma unroll
    for (int r = 0; r < 8; ++r) { bestS[r] = 3.4e38f; bestI[r] = 0x7fffffff; }

    // software-pipelined dnorm: issue next chunk's value before the WMMA block
    float dnCur = dnorm[wcol + l15];

    for (int j = 0; j < NCHUNK; ++j) {
        const int nc = j * NT;

        // ---- stage dict chunk [256 x 64] into LDS column-major -------------
        if (useT) {
            // dictT rows are contiguous: float4 loads + aligned b128 LDS stores
            #pragma unroll 4
            for (int q4 = 0; q4 < (NT * DIM) / (256 * 4); ++q4) {  // 16 iters
                int f4 = tid + 256 * q4;
                int c  = f4 >> 6;                  // column within chunk
                int kv = (f4 & 63) << 2;           // k offset (float4)
                float4 v = *(const float4*)&dictT[(long)(nc + c) * DIM + kv];
                *(float4*)&Bs[c][kv] = v;
            }
        } else {
            // fallback: read row-major dict along c (float4), scatter into Bs
            #pragma unroll 4
            for (int q4 = 0; q4 < (NT * DIM) / (256 * 4); ++q4) {  // 16 iters
                int f4 = tid + 256 * q4;
                int k  = f4 >> 4;
                int c  = (f4 & 15) << 2;
                float4 v = *(const float4*)&dict[(long)k * KDIC + nc + c];
                Bs[c + 0][k] = v.x;
                Bs[c + 1][k] = v.y;
                Bs[c + 2][k] = v.z;
                Bs[c + 3][k] = v.w;
            }
        }
        __syncthreads();

        // prefetch next chunk's dnorm so its latency hides under the WMMAs
        float dnNext = (j + 1 < NCHUNK) ? dnorm[(j + 1) * NT + wcol + l15] : 0.f;

        // ---- 16x16 similarity tile: 64 x v_wmma_f32_16x16x4_f32 ------------
        v8f acc = {0.f, 0.f, 0.f, 0.f, 0.f, 0.f, 0.f, 0.f};
        const float* arow = &As[wrow + l15][0];
        const float* brow = &Bs[wcol + l15][0];
        #pragma unroll 8
        for (int kk = 0; kk < DIM; kk += 4) {
            // A 16x4 f32 frag: lanes 0-15 hold K={0,1}, lanes 16-31 K={2,3}
            v2f a = *(const v2f*)(arow + kk + 2 * half);
            // B 4x16 f32 frag: mirrored layout from the column-major LDS tile
            v2f b = *(const v2f*)(brow + kk + 2 * half);
            acc = __builtin_amdgcn_wmma_f32_16x16x4_f32(
                      false, a, false, b, (short)0, acc, false, false);
        }

        // ---- fold into running argmin (score = ||d||^2 - 2*sim) ------------
        int nidx = nc + wcol + l15;
        #pragma unroll
        for (int r = 0; r < 8; ++r) {
            float s = dnCur - 2.0f * acc[r];
            // strict < keeps the earliest (lowest) index on exact ties,
            // matching jnp.argmin, since nidx grows with nc for a fixed lane
            if (s < bestS[r]) { bestS[r] = s; bestI[r] = nidx; }
        }
        dnCur = dnNext;
        __syncthreads();   // protect Bs before next chunk overwrites it
    }

    // ---- reduce across the 16 lanes of each half (xor butterfly) -----------
    #pragma unroll
    for (int r = 0; r < 8; ++r) {
        float s = bestS[r];
        int   i = bestI[r];
        #pragma unroll
        for (int off = 1; off < 16; off <<= 1) {
            float os = __shfl_xor(s, off, 32);
            int   oi = __shfl_xor(i, off, 32);
            if (os < s || (os == s && oi < i)) { s = os; i = oi; }
        }
        if (l15 == 0) {
            int row = wrow + r + 8 * half;
            candS[row][wave & 3] = s;
            candI[row][wave & 3] = i;
        }
    }
    __syncthreads();

    // ---- merge the 4 column-wave candidates per row -----------------------
    if (tid < MT) {
        float s = candS[tid][0];
        int   i = candI[tid][0];
        #pragma unroll
        for (int c = 1; c < 4; ++c) {
            float os = candS[tid][c];
            int   oi = candI[tid][c];
            if (os < s || (os == s && oi < i)) { s = os; i = oi; }
        }
        selIdx[tid] = i;
    }
    __syncthreads();

    // ---- gather q = 0.5*dictT[idx][:], write output, accumulate loss ------
    float lsum = 0.f;
    if (useT) {
        #pragma unroll 2
        for (int q4 = 0; q4 < 8; ++q4) {
            int f4  = tid + 256 * q4;          // float4 index, 2048 total
            int row = f4 >> 6;
            int dv  = (f4 & 63) << 2;
            int ci  = selIdx[row];
            float4 qv = *(const float4*)&dictT[(long)ci * DIM + dv];  // coalesced
            float4 xv = *(const float4*)&As[row][dv];
            float4 qq = make_float4(0.5f * qv.x, 0.5f * qv.y, 0.5f * qv.z, 0.5f * qv.w);
            float dx = xv.x - qq.x, dy = xv.y - qq.y, dz = xv.z - qq.z, dw = xv.w - qq.w;
            lsum += dx * dx + dy * dy + dz * dz + dw * dw;
            *(float4*)&out[(m0 + row) * (long)DIM + dv] = qq;         // coalesced
        }
    } else {
        #pragma unroll 4
        for (int jj = 0; jj < 32; ++jj) {
            int f   = tid + 256 * jj;
            int row = f >> 8;
            int d   = f & (DIM - 1);
            int ci  = selIdx[row];
            float q  = 0.5f * dict[(long)d * KDIC + ci];  // strided (L2) gather
            float df = As[row][d] - q;
            lsum += df * df;
            out[(m0 + row) * (long)DIM + d] = q;
        }
    }
    red[tid] = lsum;
    __syncthreads();
    #pragma unroll
    for (int s2 = 128; s2 > 0; s2 >>= 1) {
        if (tid < s2) red[tid] += red[tid + s2];
        __syncthreads();
    }
    if (tid == 0) lossPartial[blockIdx.x] = red[0];
}

// ---------------------------------------------------------------------------
// Kernel 3: final loss reduction. loss = 1.25 * sum / numel
// ---------------------------------------------------------------------------
__global__ __launch_bounds__(256) void vq_final(const float* __restrict__ partial,
                                                float* __restrict__ lossOut,
                                                int nPart, float scale) {
    __shared__ float red[256];
    float s = 0.f;
    for (int i = threadIdx.x; i < nPart; i += 256) s += partial[i];
    red[threadIdx.x] = s;
    __syncthreads();
    #pragma unroll
    for (int s2 = 128; s2 > 0; s2 >>= 1) {
        if (threadIdx.x < s2) red[threadIdx.x] += red[threadIdx.x + s2];
        __syncthreads();
    }
    if (threadIdx.x == 0) lossOut[0] = red[0] * scale;
}

// ---------------------------------------------------------------------------
extern "C" void kernel_launch(void* const* d_in, const int* in_sizes, int n_in,
                              void* d_out, int out_size, void* d_ws, size_t ws_size,
                              hipStream_t stream) {
    const float* x    = (const float*)d_in[0];   // (16,64,64,256) f32
    const float* dict = (const float*)d_in[1];   // (256,1024) f32
    float* out = (float*)d_out;                  // 65536*256 q values + 1 loss

    const long M    = 16L * 64 * 64;             // 65536 rows
    const int  nBlk = (int)(M / MT);             // 2048 workgroups

    float* ws          = (float*)d_ws;
    float* lossPartial = ws;                     // [2048]
    float* dnorm       = ws + nBlk;              // [1024]
    float* dictT       = ws + nBlk + KDIC;       // [1024*256] if workspace allows

    const size_t needT = (size_t)(nBlk + KDIC + KDIC * DIM) * sizeof(float);
    const bool haveT = (ws_size >= needT);
    if (haveT) {
        vq_transpose<<<dim3(KDIC / 32, DIM / 32), 256, 0, stream>>>(dict, dictT);
    }
    vq_norms<<<KDIC / 256, 256, 0, stream>>>(dict, dnorm);
    vq_main <<<nBlk, 256, 0, stream>>>(x, dict, haveT ? dictT : nullptr,
                                       dnorm, out, lossPartial);
    vq_final<<<1, 256, 0, stream>>>(lossPartial, out + M * DIM, nBlk,
                                    1.25f / (float)(M * DIM));
}